// kNN_23759759081725
// MI455X (gfx1250) — compile-verified
//
#include <hip/hip_runtime.h>
#include <stdint.h>

// Row-wise smallest-32 selection (kNN edges) for MI455X / gfx1250.
// One 256-thread workgroup per row; 2D grid (x = row-in-batch, y = batch).
// Row (16 KB) is staged into LDS via the CDNA5 async global->LDS path
// (ASYNCcnt, non-temporal hint since D streams exactly once vs 192 MB L2),
// selection via LDS histogram (ds_add_u32) + wave32 shuffle argmin.
//
// Assumes N is a multiple of 1024 (reference: N = 4096).

#define KSEL 32
#define TPB  256
#define NMAX 4096

__global__ __launch_bounds__(TPB)
void knn_topk_rows(const float* __restrict__ D,
                   const float* __restrict__ mask,
                   float* __restrict__ out,
                   int N, int rows) {
    __shared__ uint32_t            sKey[NMAX];     // 16 KB: raw row, then key bits
    __shared__ uint32_t            sHist[2048];    // 8 KB histogram (top 11 key bits)
    __shared__ uint32_t            sBsum[TPB];
    __shared__ unsigned long long  sWred[TPB / 32];
    __shared__ uint32_t            sWinKey[KSEL];
    __shared__ uint32_t            sWinIdx[KSEL];
    __shared__ uint32_t            sQ, sCB, sCnt;
    __shared__ unsigned long long  sLast;

    const int i = blockIdx.x;          // row within batch
    const int b = blockIdx.y;          // batch
    const int t = threadIdx.x;
    const int r = b * N + i;           // global row id in [0, B*N)

    const float* rowD = D + (size_t)r * N;
    const float* rowM = mask + (size_t)b * N;
    const float  mask_i = rowM[i];

    const int elemsPerThread = N / TPB;            // 16
    const int vecsPerThread  = elemsPerThread / 4; // 4 x b128

    // ---- Phase 0: async global -> LDS copy of the row (CDNA5 ASYNC path) ----
    {
        uint32_t ldsBase = (uint32_t)(uintptr_t)(&sKey[0]); // low 32 bits = LDS offset
        unsigned long long gbase = (unsigned long long)(uintptr_t)rowD;
        for (int k = 0; k < vecsPerThread; ++k) {
            uint32_t byteOff = (uint32_t)(t * 16 + k * (TPB * 16));
            uint32_t ldsOff  = ldsBase + byteOff;
            asm volatile("global_load_async_to_lds_b128 %0, %1, %2 th:TH_LOAD_NT"
                         :: "v"(ldsOff), "v"(byteOff), "s"(gbase)
                         : "memory");
        }
        asm volatile("s_wait_asynccnt 0x0" ::: "memory");
    }
    __syncthreads();

    // ---- Phase 1: masked sortable keys, in place ----
    // values are >= 0 so float order == uint bit order; masked -> FLT_MAX bits
    for (int e = 0; e < elemsPerThread; ++e) {
        int j = t + TPB * e;
        float d  = __uint_as_float(sKey[j]);
        float mf = mask_i * rowM[j];           // 0.0 or 1.0 exactly
        float v  = (mf != 0.0f) ? d : 3.4028234663852886e38f; // FLT_MAX
        sKey[j]  = __float_as_uint(v);
    }
    for (int k = 0; k < 2048 / TPB; ++k) sHist[t * (2048 / TPB) + k] = 0;
    if (t == 0) sCnt = 0;
    __syncthreads();

    // ---- Phase 2: histogram on top 11 bits ----
    for (int e = 0; e < elemsPerThread; ++e) {
        int j = t + TPB * e;
        atomicAdd(&sHist[sKey[j] >> 21], 1u);
    }
    __syncthreads();

    // ---- Phase 3: locate threshold bin q and count strictly below it ----
    {
        uint32_t bs = 0;
        for (int k = 0; k < 8; ++k) bs += sHist[t * 8 + k];
        sBsum[t] = bs;
    }
    __syncthreads();
    if (t == 0) {
        uint32_t c = 0, q = 0, cb = 0;
        for (int blk = 0; blk < TPB; ++blk) {
            if (c + sBsum[blk] >= KSEL) {
                uint32_t cc = c;
                for (int k = 0; k < 8; ++k) {
                    uint32_t h = sHist[blk * 8 + k];
                    if (cc + h >= KSEL) { q = blk * 8 + k; cb = cc; break; }
                    cc += h;
                }
                break;
            }
            c += sBsum[blk];
        }
        sQ = q; sCB = cb;
    }
    __syncthreads();
    const uint32_t q = sQ, cbelow = sCB;

    // ---- Phase 4: collect definite winners (bin < q); count == cbelow <= 31 ----
    for (int e = 0; e < elemsPerThread; ++e) {
        int j = t + TPB * e;
        uint32_t kk = sKey[j];
        if ((kk >> 21) < q) {
            uint32_t pos = atomicAdd(&sCnt, 1u);
            sWinKey[pos] = kk;
            sWinIdx[pos] = (uint32_t)j;
        }
    }
    __syncthreads();

    // ---- Phase 5: extract the remaining (32 - cbelow) from bin q, ascending
    //      by 64-bit key (valbits, index) -> matches lax.top_k tie stability ----
    const int nsel = KSEL - (int)cbelow;
    for (int it = 0; it < nsel; ++it) {
        unsigned long long lb  = (it == 0) ? 0ULL : sLast;
        unsigned long long loc = ~0ULL;
        for (int e = 0; e < elemsPerThread; ++e) {
            int j = t + TPB * e;
            uint32_t kk = sKey[j];
            if ((kk >> 21) == q) {
                unsigned long long c64 =
                    ((unsigned long long)kk << 32) | (uint32_t)j;
                if (it == 0 || c64 > lb) loc = (c64 < loc) ? c64 : loc;
            }
        }
        // wave32 shuffle argmin
        for (int off = 16; off > 0; off >>= 1) {
            unsigned long long o = __shfl_down(loc, off, 32);
            loc = (o < loc) ? o : loc;
        }
        if ((t & 31) == 0) sWred[t >> 5] = loc;
        __syncthreads();
        if (t == 0) {
            unsigned long long m = sWred[0];
            for (int w = 1; w < TPB / 32; ++w) m = (sWred[w] < m) ? sWred[w] : m;
            sWinKey[cbelow + it] = (uint32_t)(m >> 32);
            sWinIdx[cbelow + it] = (uint32_t)m;
            sLast = m;
        }
        __syncthreads();
    }

    // ---- Phase 6: rank the 32 winners (all 64-bit keys distinct) and emit ----
    const size_t BN32 = (size_t)rows * KSEL;
    if (t < KSEL) {
        unsigned long long my =
            ((unsigned long long)sWinKey[t] << 32) | sWinIdx[t];
        int rank = 0;
        for (int m = 0; m < KSEL; ++m) {
            unsigned long long om =
                ((unsigned long long)sWinKey[m] << 32) | sWinIdx[m];
            rank += (om < my) ? 1 : 0;
        }
        size_t   o    = (size_t)r * KSEL + (size_t)rank;
        uint32_t widx = sWinIdx[t];
        out[o]            = (float)widx;                    // edge_idx
        out[BN32 + o]     = __uint_as_float(sWinKey[t]);    // edge_D
        out[2 * BN32 + o] = mask_i * rowM[widx];            // mask_ij
    }
}

extern "C" void kernel_launch(void* const* d_in, const int* in_sizes, int n_in,
                              void* d_out, int out_size, void* d_ws, size_t ws_size,
                              hipStream_t stream) {
    (void)n_in; (void)out_size; (void)d_ws; (void)ws_size;
    const float* D    = (const float*)d_in[0];   // [B, N, N]
    const float* mask = (const float*)d_in[1];   // [B, N]
    float*       out  = (float*)d_out;           // idx | val | mask_ij, each rows*32

    const int rows = in_sizes[1];                // B*N
    const int N    = in_sizes[0] / rows;         // row length
    const int B    = rows / N;                   // batches

    dim3 grid((unsigned)N, (unsigned)B, 1);
    knn_topk_rows<<<grid, TPB, 0, stream>>>(D, mask, out, N, rows);
}